// ExLoss_13761075216688
// MI455X (gfx1250) — compile-verified
//
#include <hip/hip_runtime.h>
#include <hip/hip_bf16.h>
#include <math.h>

typedef __attribute__((ext_vector_type(2))) float v2f;
typedef __attribute__((ext_vector_type(8))) float v8f;

#define Bm 64
#define Dk 256
#define Cn 100000
#define Pp 8
#define Nn 32
#define TILE_N 128         /* block output tile: 64 x 128 */
#define KB 64              /* K staged per LDS block       */
#define LDST 68            /* 64 + 4 pad -> bank stride 4  */

// ---------------------------------------------------------------------------
// Async copy of 16 bytes global -> LDS (GLOBAL_LOAD_ASYNC_TO_LDS_B128, ASYNCcnt)
// lds_off: byte offset within the workgroup's LDS allocation (low 32 bits of
// the generic flat address of a __shared__ object; LDS aperture is the high word).
// ---------------------------------------------------------------------------
__device__ __forceinline__ void async_copy16(unsigned lds_off, const void* gptr) {
    asm volatile("global_load_async_to_lds_b128 %0, %1, off"
                 :: "v"(lds_off), "v"(gptr)
                 : "memory");
}
__device__ __forceinline__ void wait_async0() {
    asm volatile("s_wait_asynccnt 0" ::: "memory");
}

// ---------------------------------------------------------------------------
// Kernel 1: logits = inputs @ V^T  via V_WMMA_F32_16X16X4_F32
// grid: ceil(C/128) blocks, 256 threads = 8 wave32 (2 x 4 wave tiling).
// Each wave: 32x32 tile = 2x2 accumulators -> 4-deep independent WMMA chains,
// 1.0 LDS fragment loads per WMMA (2 A-frags + 2 B-frags feed 4 WMMAs).
// ---------------------------------------------------------------------------
__global__ __launch_bounds__(256) void gemm_logits(const float* __restrict__ inp,
                                                   const float* __restrict__ V,
                                                   float* __restrict__ logits) {
    __shared__ float sA[Bm * LDST];      // 64 x 68 f32  = 17408 B
    __shared__ float sB[TILE_N * LDST];  // 128 x 68 f32 = 34816 B

    const int tid  = threadIdx.x;
    const int lane = tid & 31;
    const int w    = tid >> 5;
    const int wm   = w & 1;    // 0..1 -> 32-row slice of M=64
    const int wn   = w >> 1;   // 0..3 -> 32-col slice of N=128
    const int c0   = blockIdx.x * TILE_N;

    v8f acc00 = {}, acc01 = {}, acc10 = {}, acc11 = {};

    // Per-lane fragment addressing (ISA 7.12.2, f32 16x4 A / 4x16 B layouts):
    // each fragment is 2 consecutive K floats at k0 + 2*(lane>>4).
    const int kk = 2 * (lane >> 4);
    const float* pa0 = &sA[(wm * 32 + (lane & 15)) * LDST + kk];
    const float* pa1 = &sA[(wm * 32 + 16 + (lane & 15)) * LDST + kk];
    const float* pb0 = &sB[(wn * 32 + (lane & 15)) * LDST + kk];
    const float* pb1 = &sB[(wn * 32 + 16 + (lane & 15)) * LDST + kk];

    for (int kb = 0; kb < Dk; kb += KB) {
        // Stage A block: 64 rows x 64 floats = 1024 float4, 4 per thread (async)
        for (int i = tid; i < (Bm * KB / 4); i += 256) {
            const int row = i >> 4;          // 16 float4 per row
            const int c4  = i & 15;
            async_copy16((unsigned)(size_t)&sA[row * LDST + c4 * 4],
                         inp + row * Dk + kb + c4 * 4);
        }
        // Stage B block: 128 V-rows x 64 floats = 2048 float4, 8 per thread
        for (int i = tid; i < (TILE_N * KB / 4); i += 256) {
            const int row = i >> 4;
            const int c4  = i & 15;
            int vr = c0 + row;
            if (vr >= Cn) vr = Cn - 1;       // tail rows clamped; stores guarded
            async_copy16((unsigned)(size_t)&sB[row * LDST + c4 * 4],
                         V + (size_t)vr * Dk + kb + c4 * 4);
        }
        wait_async0();
        __syncthreads();

#pragma unroll
        for (int k0 = 0; k0 < KB; k0 += 4) {
            v2f a0 = *(const v2f*)(pa0 + k0);
            v2f a1 = *(const v2f*)(pa1 + k0);
            v2f b0 = *(const v2f*)(pb0 + k0);
            v2f b1 = *(const v2f*)(pb1 + k0);
            acc00 = __builtin_amdgcn_wmma_f32_16x16x4_f32(
                false, a0, false, b0, (short)0, acc00, false, false);
            acc01 = __builtin_amdgcn_wmma_f32_16x16x4_f32(
                false, a0, false, b1, (short)0, acc01, false, false);
            acc10 = __builtin_amdgcn_wmma_f32_16x16x4_f32(
                false, a1, false, b0, (short)0, acc10, false, false);
            acc11 = __builtin_amdgcn_wmma_f32_16x16x4_f32(
                false, a1, false, b1, (short)0, acc11, false, false);
        }
        __syncthreads();
    }

    // Store four 16x16 f32 D tiles: VGPR r -> (M = r + 8*(lane>=16), N = lane&15)
    const int n0 = c0 + wn * 32 + (lane & 15);
    const int n1 = n0 + 16;
    const int m0 = wm * 32 + 8 * (lane >> 4);
    const int m1 = m0 + 16;
#pragma unroll
    for (int r = 0; r < 8; ++r) {
        if (n0 < Cn) {
            logits[(size_t)(m0 + r) * Cn + n0] = acc00[r];
            logits[(size_t)(m1 + r) * Cn + n0] = acc10[r];
        }
        if (n1 < Cn) {
            logits[(size_t)(m0 + r) * Cn + n1] = acc01[r];
            logits[(size_t)(m1 + r) * Cn + n1] = acc11[r];
        }
    }
}

// ---------------------------------------------------------------------------
// Kernel 2: per-row streaming logsumexp over C=100000 (float4 reads);
// bu_row[b] = lse - logit[target]
// ---------------------------------------------------------------------------
__global__ __launch_bounds__(256) void rowlse(const float* __restrict__ logits,
                                              const int* __restrict__ targets,
                                              float* __restrict__ bu_row) {
    __shared__ float sm[256];
    __shared__ float ss[256];
    const int b = blockIdx.x;
    const float* row = logits + (size_t)b * Cn;
    const float4* row4 = (const float4*)row;

    float m = -INFINITY, s = 0.0f;
    for (int c4 = threadIdx.x; c4 < (Cn / 4); c4 += 256) {
        const float4 v = row4[c4];
        const float lm = fmaxf(fmaxf(v.x, v.y), fmaxf(v.z, v.w));
        const float nm = fmaxf(m, lm);
        s = s * __expf(m - nm)
          + __expf(v.x - nm) + __expf(v.y - nm)
          + __expf(v.z - nm) + __expf(v.w - nm);
        m = nm;
    }
    sm[threadIdx.x] = m;
    ss[threadIdx.x] = s;
    __syncthreads();
    for (int off = 128; off > 0; off >>= 1) {
        if (threadIdx.x < off) {
            const float m1 = sm[threadIdx.x], s1 = ss[threadIdx.x];
            const float m2 = sm[threadIdx.x + off], s2 = ss[threadIdx.x + off];
            const float nm = fmaxf(m1, m2);
            sm[threadIdx.x] = nm;
            ss[threadIdx.x] = s1 * __expf(m1 - nm) + s2 * __expf(m2 - nm);
        }
        __syncthreads();
    }
    if (threadIdx.x == 0) {
        const float lse = sm[0] + __logf(ss[0]);
        bu_row[b] = lse - row[targets[b]];
    }
}

// ---------------------------------------------------------------------------
// Kernel 3: cosine sim among batch, hardest-pos / hard-neg losses, final mean.
// ---------------------------------------------------------------------------
__device__ __forceinline__ float softplus_f(float x) {
    return fmaxf(x, 0.0f) + log1pf(__expf(-fabsf(x)));
}

__global__ __launch_bounds__(256) void simloss(const float* __restrict__ inp,
                                               const int* __restrict__ pos_idx,
                                               const int* __restrict__ pos_mask,
                                               const int* __restrict__ neg_idx,
                                               const int* __restrict__ neg_mask,
                                               const float* __restrict__ bu_row,
                                               float* __restrict__ out) {
    __shared__ float nrm[Bm];
    __shared__ float sim[Bm * 65];
    __shared__ float red[256];
    const int tid = threadIdx.x;

    if (tid < Bm) {
        float s = 0.0f;
        const float* r = inp + tid * Dk;
        for (int k = 0; k < Dk; ++k) s += r[k] * r[k];
        nrm[tid] = fmaxf(sqrtf(s), 1e-12f);
    }
    __syncthreads();

    for (int p = tid; p < Bm * Bm; p += 256) {
        const int i = p >> 6, j = p & 63;
        const float* ri = inp + i * Dk;
        const float* rj = inp + j * Dk;
        float s = 0.0f;
        for (int k = 0; k < Dk; ++k) s += ri[k] * rj[k];
        sim[i * 65 + j] = s / (nrm[i] * nrm[j]);
    }
    __syncthreads();

    float hrow = 0.0f;
    if (tid < Bm) {
        const int b = tid;
        float hp = INFINITY;
        for (int p = 0; p < Pp; ++p) {
            if (pos_mask[b * Pp + p]) {
                hp = fminf(hp, sim[b * 65 + pos_idx[b * Pp + p]]);
            }
        }
        const float hp_loss = softplus_f(-hp);
        const float thr = hp - 0.3f;
        int cnt = 0;
        float hsum = 0.0f;
        for (int n = 0; n < Nn; ++n) {
            if (neg_mask[b * Nn + n]) {
                const float s = sim[b * 65 + neg_idx[b * Nn + n]];
                if (s > thr) { ++cnt; hsum += softplus_f(s); }
            }
        }
        const float hn_loss = (cnt > 0) ? (hsum / (float)cnt) : 0.0f;
        hrow = hp_loss + hn_loss + bu_row[b];
    }
    red[tid] = (tid < Bm) ? hrow : 0.0f;
    __syncthreads();
    for (int off = 128; off > 0; off >>= 1) {
        if (tid < off) red[tid] += red[tid + off];
        __syncthreads();
    }
    if (tid == 0) out[0] = red[0] / (float)Bm;
}

// ---------------------------------------------------------------------------
extern "C" void kernel_launch(void* const* d_in, const int* in_sizes, int n_in,
                              void* d_out, int out_size, void* d_ws, size_t ws_size,
                              hipStream_t stream) {
    const float* inputs   = (const float*)d_in[0];
    const float* V        = (const float*)d_in[1];
    const int*   targets  = (const int*)d_in[2];
    const int*   pos_idx  = (const int*)d_in[3];
    const int*   pos_mask = (const int*)d_in[4];
    const int*   neg_idx  = (const int*)d_in[5];
    const int*   neg_mask = (const int*)d_in[6];

    float* out    = (float*)d_out;
    float* logits = out + 1;            // out[0] = loss, out[1..] = logits [64 x 100000]
    float* bu_row = (float*)d_ws;       // 64 floats of scratch

    const int nblocks = (Cn + TILE_N - 1) / TILE_N;   // 782
    gemm_logits<<<nblocks, 256, 0, stream>>>(inputs, V, logits);
    rowlse<<<Bm, 256, 0, stream>>>(logits, targets, bu_row);
    simloss<<<1, 256, 0, stream>>>(inputs, pos_idx, pos_mask, neg_idx, neg_mask,
                                   bu_row, out);
}